// NDCG_9139690405902
// MI455X (gfx1250) — compile-verified
//
#include <hip/hip_runtime.h>
#include <float.h>

typedef float v2f __attribute__((ext_vector_type(2)));
typedef float v8f __attribute__((ext_vector_type(8)));

#define BB    16384   // batch
#define NCLS  10      // classes
#define TABP  16400   // padded AB/hist table length (multiple of 16 above BB)
#define DLEN  16448   // discount table length (covers P+N+padding)

// workspace layout, in floats
#define OFF_DTAB   0
#define OFF_SPLUS  (OFF_DTAB + DLEN)            // NCLS*BB
#define OFF_SMINUS (OFF_SPLUS + NCLS*BB)        // NCLS*BB
#define OFF_ABTAB  (OFF_SMINUS + NCLS*BB)       // NCLS*TABP*2 (interleaved A,B pairs)
#define OFF_OPT    (OFF_ABTAB + NCLS*TABP*2)    // NCLS*BB
#define OFF_C      (OFF_OPT + NCLS*BB)          // NCLS
#define OFF_LOSS   (OFF_C + NCLS)               // NCLS
#define OFF_P      (OFF_LOSS + NCLS)            // NCLS (int)
#define OFF_CNT    (OFF_P + NCLS)               // NCLS (int)
#define OFF_HIST   (OFF_CNT + NCLS)             // NCLS*TABP (int)

// ---------------- k0: discount table + counter init ----------------
__global__ void k0_init(float* F) {
  int i = blockIdx.x * blockDim.x + threadIdx.x;
  if (i < DLEN) F[OFF_DTAB + i] = (i == 0) ? 0.f : 1.f / log2f((float)(i + 1));
  if (i < NCLS) {
    ((int*)(F + OFF_CNT))[i] = 0;
    ((int*)(F + OFF_P))[i]   = 0;
    F[OFF_LOSS + i] = 0.f;
    F[OFF_C + i]    = 1.f;
  }
}

// ---------------- k1: per-class partition + bitonic sort in LDS ----------------
__device__ void bitonic_sort(float* buf, int t, int nt) {
  for (unsigned k = 2; k <= BB; k <<= 1) {
    for (unsigned j = k >> 1; j > 0; j >>= 1) {
      for (unsigned i = (unsigned)t; i < BB; i += nt) {
        unsigned ixj = i ^ j;
        if (ixj > i) {
          float a = buf[i], b = buf[ixj];
          bool sw = ((i & k) == 0) ? (a > b) : (a < b);
          if (sw) { buf[i] = b; buf[ixj] = a; }
        }
      }
      __syncthreads();
    }
  }
}

__global__ __launch_bounds__(1024) void k1_sort(const float* __restrict__ scores,
                                                const int* __restrict__ labels,
                                                float* F) {
  __shared__ float buf[BB];   // 64KB LDS (320KB/WGP on gfx1250)
  const int c = blockIdx.x, t = threadIdx.x, nt = blockDim.x;
  int* cnt = (int*)(F + OFF_CNT) + c;
  int* Pp  = (int*)(F + OFF_P) + c;

  // phase A: positives. Store -s, sort ascending -> s_plus descending.
  for (int i = t; i < BB; i += nt)
    if (labels[i] == c) { int idx = atomicAdd(cnt, 1); buf[idx] = -scores[i * NCLS + c]; }
  __threadfence(); __syncthreads();
  const int P = atomicAdd(cnt, 0);
  for (int i = P + t; i < BB; i += nt) buf[i] = FLT_MAX;   // sentinels sort last
  __syncthreads();
  bitonic_sort(buf, t, nt);
  for (int i = t; i < BB; i += nt) F[OFF_SPLUS + c * BB + i] = -buf[i];
  __syncthreads();
  if (t == 0) { *Pp = P; atomicExch(cnt, 0); }
  __threadfence(); __syncthreads();

  // phase B: negatives ascending.
  for (int i = t; i < BB; i += nt)
    if (labels[i] != c) { int idx = atomicAdd(cnt, 1); buf[idx] = scores[i * NCLS + c]; }
  __threadfence(); __syncthreads();
  const int N = atomicAdd(cnt, 0);
  for (int i = N + t; i < BB; i += nt) buf[i] = FLT_MAX;
  __syncthreads();
  bitonic_sort(buf, t, nt);
  for (int i = t; i < BB; i += nt) F[OFF_SMINUS + c * BB + i] = buf[i];
}

// ---------------- k2: AB tables (Cconst pre-folded) + Cconst ----------------
// argmax_l is invariant under scaling by Cconst>0, so store A'=C*A, B'=C*B and
// feed RAW dTab into the WMMA C operand (no per-iteration invC multiply).
__global__ __launch_bounds__(1024) void k2_tables(float* F) {
  __shared__ float sc[1024];
  __shared__ float red[1024];
  const int c = blockIdx.x, t = threadIdx.x;
  int* histc = (int*)(F + OFF_HIST) + c * TABP;
  for (int i = t; i < TABP; i += 1024) histc[i] = 0;   // zero hist for k4
  const int P = ((int*)(F + OFF_P))[c];
  const int N = BB - P;
  if (P == 0 || N == 0) return;
  const float* sp = F + OFF_SPLUS + c * BB;
  float* ABt = F + OFF_ABTAB + c * TABP * 2;
  const float pn = (float)P * (float)N;

  // Cconst = sum_{i=1..P} D(i)  (needed as the fold-in scale below)
  float cc = 0.f;
  for (int i = 1 + t; i <= P; i += 1024) cc += F[OFF_DTAB + i];
  red[t] = cc; __syncthreads();
  for (int s = 512; s > 0; s >>= 1) { if (t < s) red[t] += red[t + s]; __syncthreads(); }
  const float Cc = red[0]; __syncthreads();
  if (t == 0) F[OFF_C + c] = Cc;

  // chunk-of-16 partial sums over s_plus[0..P)
  const int base = t * 16;
  float cs = 0.f;
  for (int j = 0; j < 16; ++j) { int k = base + j; if (k < P) cs += sp[k]; }
  sc[t] = cs; __syncthreads();
  for (int off = 1; off < 1024; off <<= 1) {     // Hillis-Steele inclusive scan
    float v = (t >= off) ? sc[t - off] : 0.f; __syncthreads();
    sc[t] += v; __syncthreads();
  }
  const float total = sc[1023];
  float run = (t > 0) ? sc[t - 1] : 0.f;         // exclusive prefix base
  const float sA = Cc * (-2.f / pn);
  const float sB = Cc * ( 2.f / pn);
  for (int j = 0; j < 16; ++j) {
    int k = base + j;                            // k = l-1, valid for k<=P
    if (k <= P) {
      ABt[2 * k]     = sA * (total - run);       // Cc * -2/(PN) * Sl[k]
      ABt[2 * k + 1] = sB * (float)(P - k);      // Cc *  2/(PN) * cnt[k]
      if (k < P) run += sp[k];
    }
  }
  for (int k = P + 1 + t; k < TABP; k += 1024) { ABt[2 * k] = -1e30f; ABt[2 * k + 1] = 0.f; }
}

// ---------------- k3: WMMA argmax over opt_vals [N x (P+1)] ----------------
// opt_vals tile (scaled by Cconst) = A(16x4: rows (1, s_minus[m], 0, 0))
//   x B(4x16: cols (A'[l],B'[l],*,*)) + C(Hankel: dTab[l+m])
// One v_wmma_f32_16x16x4_f32 per 16x16 tile. B rows K=2,3 may hold arbitrary
// finite data (duplicated table values) because A columns K=2,3 are exactly 0.
__global__ __launch_bounds__(256) void k3_argmax(float* F) {
  const int c = blockIdx.y;
  const int P = ((int*)(F + OFF_P))[c];
  const int N = BB - P;
  if (P == 0 || N == 0) return;
  const int wave = threadIdx.x >> 5;
  const int lane = threadIdx.x & 31;
  const int m0 = (blockIdx.x * 8 + wave) * 16;     // 16 m-rows per wave
  if (m0 >= N) return;                              // wave-uniform exit
  const float* __restrict__ ABt = F + OFF_ABTAB + c * TABP * 2;
  const float* __restrict__ dT  = F + OFF_DTAB;
  const float* __restrict__ smv = F + OFF_SMINUS + c * BB;

  // A operand (16x4 f32): lanes 0-15 hold K=0,1 for row M=lane; lanes 16-31 (K=2,3)=0
  v2f av;
  if (lane < 16) { av.x = 1.f; av.y = smv[m0 + lane]; }
  else           { av.x = 0.f; av.y = 0.f; }

  const int numL = (P + 1 + 15) >> 4;
  v8f best;
  float bestL[8];
  #pragma unroll
  for (int v = 0; v < 8; ++v) { best[v] = -3.4e38f; bestL[v] = 1.f; }

  const int li     = lane & 15;                          // N position of this lane
  const int cbase0 = m0 + 1 + li + ((lane >> 4) << 3);   // + M offset (rows 0-7 / 8-15)

  // ---- software pipeline: preload tile 0 ----
  v2f bv_cur; bv_cur.x = ABt[2 * li]; bv_cur.y = ABt[2 * li + 1];
  v8f cv_cur;
  #pragma unroll
  for (int v = 0; v < 8; ++v) cv_cur[v] = dT[cbase0 + v];

  float lvalf = (float)(li + 1);
  for (int lt = 0; lt < numL; ++lt) {
    // prefetch/load next tile's operands while the current tile computes
    const int nlt = (lt + 1 < numL) ? (lt + 1) : lt;
    const int nb  = (nlt << 4) + li;
    const int ncb = cbase0 + (nlt << 4);
    __builtin_prefetch(dT + ncb + 16, 0, 1);             // gfx1250 global_prefetch
    v2f bv_nxt; bv_nxt.x = ABt[2 * nb]; bv_nxt.y = ABt[2 * nb + 1];
    v8f cv_nxt;
    #pragma unroll
    for (int v = 0; v < 8; ++v) cv_nxt[v] = dT[ncb + v];

    v8f d = __builtin_amdgcn_wmma_f32_16x16x4_f32(false, av, false, bv_cur,
                                                  (short)0, cv_cur, false, false);
    #pragma unroll
    for (int v = 0; v < 8; ++v)
      if (d[v] > best[v]) { best[v] = d[v]; bestL[v] = lvalf; }  // strict: first max wins
    lvalf += 16.f;
    bv_cur = bv_nxt;
    cv_cur = cv_nxt;
  }

  // arg-reduce across the 16 lanes of each half-wave (rows 0-7 in lanes 0-15, 8-15 in 16-31)
  #pragma unroll
  for (int msk = 1; msk < 16; msk <<= 1) {
    #pragma unroll
    for (int v = 0; v < 8; ++v) {
      float ov = __shfl_xor(best[v], msk, 32);
      float ol = __shfl_xor(bestL[v], msk, 32);
      if (ov > best[v] || (ov == best[v] && ol < bestL[v])) { best[v] = ov; bestL[v] = ol; }
    }
  }
  float* opt = F + OFF_OPT + c * BB;
  if (lane == 0) {
    #pragma unroll
    for (int v = 0; v < 8; ++v) if (m0 + v < N) opt[m0 + v] = bestL[v];
  }
  if (lane == 16) {
    #pragma unroll
    for (int v = 0; v < 8; ++v) if (m0 + 8 + v < N) opt[m0 + 8 + v] = bestL[v];
  }
}

// ---------------- k4: histogram -> r_plus -> per-class loss ----------------
__device__ float bred(float v, float* red, int t) {
  red[t] = v; __syncthreads();
  for (int s = 512; s > 0; s >>= 1) { if (t < s) red[t] += red[t + s]; __syncthreads(); }
  float r = red[0]; __syncthreads();
  return r;
}

__global__ __launch_bounds__(1024) void k4_final(float* F) {
  __shared__ int   isc[1024];
  __shared__ float red[1024];
  const int c = blockIdx.x, t = threadIdx.x;
  const int P = ((int*)(F + OFF_P))[c];
  const int N = BB - P;
  if (P == 0 || N == 0) { if (t == 0) F[OFF_LOSS + c] = 0.f; return; }
  int* histc = (int*)(F + OFF_HIST) + c * TABP;
  const float* opt = F + OFF_OPT + c * BB;
  const float* sp  = F + OFF_SPLUS + c * BB;
  const float* sm  = F + OFF_SMINUS + c * BB;
  const float* dT  = F + OFF_DTAB;

  for (int m = t; m < N; m += 1024)
    atomicAdd(&histc[(int)(opt[m] + 0.5f)], 1);
  __threadfence(); __syncthreads();

  // scan histogram over bins [1..P+1], chunks of 16
  const int base = t * 16 + 1;
  int cs = 0;
  for (int j = 0; j < 16; ++j) { int k = base + j; if (k <= P + 1) cs += histc[k]; }
  isc[t] = cs; __syncthreads();
  for (int off = 1; off < 1024; off <<= 1) {
    int v = (t >= off) ? isc[t - off] : 0; __syncthreads();
    isc[t] += v; __syncthreads();
  }
  int run = (t > 0) ? isc[t - 1] : 0;
  float dsum = 0.f, fpos = 0.f;
  for (int j = 0; j < 16; ++j) {
    int k = base + j;
    if (k <= P) {                       // i = k in 1..P ; r_plus = 1 + cum<=k
      run += histc[k];
      float r = (float)(1 + run);
      dsum += dT[k + run];              // D((i-1) + r_plus_i)
      fpos += sp[k - 1] * ((float)N + 2.f - 2.f * r);
    }
  }
  float smsum = 0.f, fneg = 0.f, spsum = 0.f;
  for (int m = t; m < N; m += 1024) {
    float s = sm[m]; smsum += s;
    fneg += s * ((float)P + 2.f - 2.f * opt[m]);
  }
  for (int i = t; i < P; i += 1024) spsum += sp[i];

  dsum  = bred(dsum,  red, t);
  fpos  = bred(fpos,  red, t);
  fneg  = bred(fneg,  red, t);
  smsum = bred(smsum, red, t);
  spsum = bred(spsum, red, t);
  if (t == 0) {
    const float C = F[OFF_C + c];
    const float Delta = 1.f - dsum / C;
    const float FR  = fpos + fneg;
    const float FRs = (float)N * spsum - (float)P * smsum;
    F[OFF_LOSS + c] = Delta + (FR - FRs) / ((float)P * (float)N);
  }
}

// ---------------- k5: deterministic ordered sum ----------------
__global__ void k5_out(const float* __restrict__ F, float* __restrict__ out) {
  float s = 0.f;
  for (int c = 0; c < NCLS; ++c) s += F[OFF_LOSS + c];
  out[0] = s / (float)NCLS;
}

extern "C" void kernel_launch(void* const* d_in, const int* in_sizes, int n_in,
                              void* d_out, int out_size, void* d_ws, size_t ws_size,
                              hipStream_t stream) {
  const float* scores = (const float*)d_in[0];   // [B, NCLS] row-major f32
  const int*   labels = (const int*)d_in[1];     // [B] int32
  float* F   = (float*)d_ws;
  float* out = (float*)d_out;

  k0_init  <<<(DLEN + 255) / 256, 256, 0, stream>>>(F);
  k1_sort  <<<NCLS, 1024, 0, stream>>>(scores, labels, F);
  k2_tables<<<NCLS, 1024, 0, stream>>>(F);
  k3_argmax<<<dim3(BB / (16 * 8), NCLS), 256, 0, stream>>>(F);
  k4_final <<<NCLS, 1024, 0, stream>>>(F);
  k5_out   <<<1, 1, 0, stream>>>(F, out);
}